// MyFastRCNNOutputLayers_23691039605237
// MI455X (gfx1250) — compile-verified
//
#include <hip/hip_runtime.h>
#include <hip/hip_bf16.h>

typedef float v2f __attribute__((ext_vector_type(2)));
typedef float v8f __attribute__((ext_vector_type(8)));

#define R_BOXES 40000
#define K_CLS   80
#define NK      (R_BOXES * K_CLS)   // 3,200,000
#define M_PRE   2048
#define TOPK    100
#define SCORE_THRESH 0.05f
#define NMS_THRESH   0.5f
#define NBINS   4096
#define CAND_CAP 8192

// ---- workspace byte offsets (zero region is contiguous: [0, 606272)) ----
#define WS_HIST   0          // u32[4096]
#define WS_META   16384      // u32[16]   [0]=cut bin, [2]=cand count
#define WS_KEYS   16448      // u64[8192]
#define WS_MASK   81984      // u32[2048*64] iou>thresh bitmask (8B aligned)
#define WS_ZERO_WORDS 151568 // (606272 bytes)/4
#define WS_TOPV   606272     // f32[2048]
#define WS_TOPI   614464     // i32[2048]
#define WS_BOFF   622656     // f32[2048*4] clipped + class offset (for IoU)
#define WS_BOUT   655424     // f32[2048*4] clipped (for output)
#define WS_AREA   688192     // f32[2048]
#define WS_CLS    696384     // i32[2048]
#define WS_VALID  704576     // u32[2048]

// ---------------------------------------------------------------- zero scratch
__global__ void zero_kernel(unsigned* p, int nwords) {
    for (int i = blockIdx.x * blockDim.x + threadIdx.x; i < nwords;
         i += gridDim.x * blockDim.x)
        p[i] = 0u;
}

// ---------------------------------------------------------------- histogram
__global__ void hist_kernel(const float* __restrict__ scores, unsigned* hist) {
    int stride = gridDim.x * blockDim.x;
    for (int idx = blockIdx.x * blockDim.x + threadIdx.x; idx < NK; idx += stride) {
        int r = idx / K_CLS;
        int c = idx - r * K_CLS;
        int soff = r * (K_CLS + 1) + c;
        if (idx + stride < NK) {
            int r2 = (idx + stride) / K_CLS;
            int c2 = (idx + stride) - r2 * K_CLS;
            __builtin_prefetch(&scores[r2 * (K_CLS + 1) + c2], 0, 0); // global_prefetch_b8
        }
        float s = scores[soff];
        if (s > SCORE_THRESH) {
            int bin = (int)(s * (float)NBINS);
            bin = bin < 0 ? 0 : (bin > NBINS - 1 ? NBINS - 1 : bin);
            atomicAdd(&hist[bin], 1u);
        }
    }
}

// ---------------------------------------------------------------- cutoff bin
__global__ void cutoff_kernel(const unsigned* __restrict__ hist, unsigned* meta) {
    if (threadIdx.x == 0) {
        unsigned acc = 0;
        int B = 0;
        for (int b = NBINS - 1; b >= 0; --b) {
            acc += hist[b];
            if (acc >= M_PRE) { B = b; break; }
        }
        meta[0] = (unsigned)B;
    }
}

// ---------------------------------------------------------------- compact survivors
__global__ void compact_kernel(const float* __restrict__ scores, unsigned* meta,
                               unsigned long long* __restrict__ keys) {
    int B = (int)meta[0];
    int stride = gridDim.x * blockDim.x;
    for (int idx = blockIdx.x * blockDim.x + threadIdx.x; idx < NK; idx += stride) {
        int r = idx / K_CLS;
        int c = idx - r * K_CLS;
        float s = scores[r * (K_CLS + 1) + c];
        if (s > SCORE_THRESH) {
            int bin = (int)(s * (float)NBINS);
            bin = bin < 0 ? 0 : (bin > NBINS - 1 ? NBINS - 1 : bin);
            if (bin >= B) {
                unsigned pos = atomicAdd(&meta[2], 1u);
                if (pos < CAND_CAP) {
                    unsigned sb = __float_as_uint(s); // s>0 => monotone as uint
                    keys[pos] = ((unsigned long long)sb << 32) | (unsigned)(~idx);
                }
            }
        }
    }
}

// ---------------------------------------------------------------- bitonic sort (desc)
__global__ __launch_bounds__(1024) void sort_kernel(unsigned long long* __restrict__ keys,
                                                    float* __restrict__ topv,
                                                    int* __restrict__ topi) {
    __shared__ unsigned long long k[CAND_CAP]; // 64 KB of the 320 KB WGP LDS
    int tid = threadIdx.x;
    for (int i = tid; i < CAND_CAP; i += 1024) k[i] = keys[i]; // zeros pad the tail
    for (int size = 2; size <= CAND_CAP; size <<= 1) {
        for (int str = size >> 1; str > 0; str >>= 1) {
            __syncthreads();
            for (int i = tid; i < CAND_CAP; i += 1024) {
                int j = i ^ str;
                if (j > i) {
                    bool desc = ((i & size) == 0);
                    unsigned long long a = k[i], b = k[j];
                    if (desc ? (a < b) : (a > b)) { k[i] = b; k[j] = a; }
                }
            }
        }
    }
    __syncthreads();
    for (int q = tid; q < M_PRE; q += 1024) {
        unsigned long long key = k[q];
        if (key != 0ull) {
            topv[q] = __uint_as_float((unsigned)(key >> 32));
            topi[q] = (int)(~(unsigned)key);
        } else {
            topv[q] = -1.0f;
            topi[q] = 0;
        }
    }
}

// ---------------------------------------------------------------- gather + clip
__global__ void gather_kernel(const float* __restrict__ boxes,
                              const float* __restrict__ topv,
                              const int* __restrict__ topi,
                              const int* __restrict__ image_h,
                              const int* __restrict__ image_w,
                              float* __restrict__ bOff, float* __restrict__ bOut,
                              float* __restrict__ area, int* __restrict__ cls,
                              unsigned* __restrict__ validf) {
    int i = blockIdx.x * blockDim.x + threadIdx.x;
    if (i >= M_PRE) return;
    float W = (float)(*image_w), H = (float)(*image_h);
    int idx = topi[i];
    int r = idx / K_CLS;
    int c = idx - r * K_CLS;
    const float* bp = boxes + r * (K_CLS * 4) + c * 4;
    float x0 = fminf(fmaxf(bp[0], 0.f), W);
    float y0 = fminf(fmaxf(bp[1], 0.f), H);
    float x1 = fminf(fmaxf(bp[2], 0.f), W);
    float y1 = fminf(fmaxf(bp[3], 0.f), H);
    float off = (fmaxf(W, H) + 1.0f) * (float)c;
    bOut[i * 4 + 0] = x0; bOut[i * 4 + 1] = y0;
    bOut[i * 4 + 2] = x1; bOut[i * 4 + 3] = y1;
    bOff[i * 4 + 0] = x0 + off; bOff[i * 4 + 1] = y0 + off;
    bOff[i * 4 + 2] = x1 + off; bOff[i * 4 + 3] = y1 + off;
    area[i]  = (x1 - x0) * (y1 - y0);
    cls[i]   = c;
    validf[i] = (topv[i] > SCORE_THRESH) ? 1u : 0u;
}

// ---------------------------------------------------------------- pairwise IoU
// One wave32 per 16x16 tile. Union denominator S[i][j]=area[i]+area[j] computed
// as a rank-2 outer product via v_wmma_f32_16x16x4_f32 (A=[area,1,0,0], B=[1;area;0;0]).
__global__ void iou_kernel(const float* __restrict__ bOff,
                           const float* __restrict__ area,
                           unsigned* __restrict__ mask) {
    int wave = (blockIdx.x * blockDim.x + threadIdx.x) >> 5; // 16384 waves
    int lane = threadIdx.x & 31;
    int tr = wave >> 7, tc = wave & 127;
    int i0 = tr * 16, j0 = tc * 16;

    v2f a, b;
    a.x = (lane < 16) ? area[i0 + lane] : 0.0f;  // A: K=0 col = area_i
    a.y = (lane < 16) ? 1.0f : 0.0f;             //    K=1 col = 1
    b.x = (lane < 16) ? 1.0f : 0.0f;             // B: K=0 row = 1
    b.y = (lane < 16) ? area[j0 + lane] : 0.0f;  //    K=1 row = area_j
    v8f cz = {};
    v8f S = __builtin_amdgcn_wmma_f32_16x16x4_f32(false, a, false, b,
                                                  (short)0, cz, false, false);

    int j = j0 + (lane & 15);
    float4 bj = ((const float4*)bOff)[j];
    int rowsel = (lane >> 4) << 3;

#pragma unroll
    for (int k = 0; k < 8; ++k) {
        int i = i0 + k + rowsel;
        float4 bi = ((const float4*)bOff)[i];
        float ltx = fmaxf(bi.x, bj.x), lty = fmaxf(bi.y, bj.y);
        float rbx = fminf(bi.z, bj.z), rby = fminf(bi.w, bj.w);
        float w = fmaxf(rbx - ltx, 0.f), h = fmaxf(rby - lty, 0.f);
        float inter = w * h;
        bool pred = inter > NMS_THRESH * (S[k] - inter); // iou > 0.5 (union>=0)
        unsigned bal = (unsigned)__ballot(pred);         // wave32: low 32 bits
        if (lane == 0) {
            unsigned lo = bal & 0xFFFFu, hi = bal >> 16;
            int word = j0 >> 5, sh = j0 & 31;            // sh is 0 or 16
            atomicOr(&mask[(i0 + k) * 64 + word], lo << sh);
            atomicOr(&mask[(i0 + 8 + k) * 64 + word], hi << sh);
        }
    }
}

// ---------------------------------------------------------------- serial NMS + output
// Single wave32: the 2048-bit suppressed set = one u64 per lane. No barriers in
// the 2048-step serial scan; owner lane's word is broadcast with a 64-bit shuffle,
// row bitmask loads are software-pipelined one row ahead (independent of the chain).
__global__ void nms_kernel(const unsigned long long* __restrict__ mask64,
                           const unsigned* __restrict__ validf,
                           const float* __restrict__ topv,
                           const float* __restrict__ bOut,
                           const int* __restrict__ cls,
                           float* __restrict__ out) {
    int lane = threadIdx.x & 31;
    unsigned long long sup = 0ull, keepbits = 0ull;
    unsigned long long rw = mask64[lane]; // row 0 (this lane's 64-bit slice)
    for (int i = 0; i < M_PRE; ++i) {
        unsigned long long nxt = (i + 1 < M_PRE) ? mask64[(i + 1) * 32 + lane] : 0ull;
        int owner = i >> 6;                         // lane owning bit i
        unsigned long long sw = __shfl(sup, owner, 32);
        bool suppressed = (sw >> (i & 63)) & 1ull;
        bool keep = (!suppressed) && (validf[i] != 0u); // uniform across wave
        if (keep) {
            sup |= rw;
            if (lane == owner) keepbits |= (1ull << (i & 63));
        }
        rw = nxt;
    }
    __shared__ unsigned long long keepb[32];
    keepb[lane] = keepbits;
    __syncthreads();
    if (lane == 0) {
        int rank = 0;
        for (int i = 0; i < M_PRE && rank < TOPK; ++i) {
            if ((keepb[i >> 6] >> (i & 63)) & 1ull) {
                out[rank * 4 + 0] = bOut[i * 4 + 0];
                out[rank * 4 + 1] = bOut[i * 4 + 1];
                out[rank * 4 + 2] = bOut[i * 4 + 2];
                out[rank * 4 + 3] = bOut[i * 4 + 3];
                out[400 + rank] = topv[i];
                ((int*)out)[500 + rank] = cls[i];
                out[600 + rank] = 1.0f;
                ++rank;
            }
        }
        for (; rank < TOPK; ++rank) {            // defaults for unfilled slots
            out[rank * 4 + 0] = 0.0f; out[rank * 4 + 1] = 0.0f;
            out[rank * 4 + 2] = 0.0f; out[rank * 4 + 3] = 0.0f;
            out[400 + rank] = 0.0f;
            ((int*)out)[500 + rank] = -1;
            out[600 + rank] = 0.0f;
        }
    }
}

// ---------------------------------------------------------------- launch
extern "C" void kernel_launch(void* const* d_in, const int* in_sizes, int n_in,
                              void* d_out, int out_size, void* d_ws, size_t ws_size,
                              hipStream_t stream) {
    const float* boxes  = (const float*)d_in[0];
    const float* scores = (const float*)d_in[1];
    const int* image_h  = (const int*)d_in[2];
    const int* image_w  = (const int*)d_in[3];
    char* ws = (char*)d_ws;

    unsigned*           hist  = (unsigned*)(ws + WS_HIST);
    unsigned*           meta  = (unsigned*)(ws + WS_META);
    unsigned long long* keys  = (unsigned long long*)(ws + WS_KEYS);
    unsigned*           mask  = (unsigned*)(ws + WS_MASK);
    float*              topv  = (float*)(ws + WS_TOPV);
    int*                topi  = (int*)(ws + WS_TOPI);
    float*              bOff  = (float*)(ws + WS_BOFF);
    float*              bOut  = (float*)(ws + WS_BOUT);
    float*              area  = (float*)(ws + WS_AREA);
    int*                cls   = (int*)(ws + WS_CLS);
    unsigned*           validf= (unsigned*)(ws + WS_VALID);
    float*              out   = (float*)d_out;

    zero_kernel<<<(WS_ZERO_WORDS + 255) / 256, 256, 0, stream>>>((unsigned*)ws, WS_ZERO_WORDS);
    hist_kernel<<<4096, 256, 0, stream>>>(scores, hist);
    cutoff_kernel<<<1, 32, 0, stream>>>(hist, meta);
    compact_kernel<<<4096, 256, 0, stream>>>(scores, meta, keys);
    sort_kernel<<<1, 1024, 0, stream>>>(keys, topv, topi);
    gather_kernel<<<(M_PRE + 255) / 256, 256, 0, stream>>>(boxes, topv, topi, image_h, image_w,
                                                           bOff, bOut, area, cls, validf);
    iou_kernel<<<2048, 256, 0, stream>>>(bOff, area, mask);
    nms_kernel<<<1, 32, 0, stream>>>((const unsigned long long*)(ws + WS_MASK),
                                     validf, topv, bOut, cls, out);
}